// QDeltaNet_27582279975482
// MI455X (gfx1250) — compile-verified
//
#include <hip/hip_runtime.h>
#include <math.h>

#define Bsz 256
#define Tsz 1024
#define INPUTD 64
#define HIDDEN 128
#define CTXD 192
#define SROW 132           // padded LDS row stride (floats) -> conflict-free WMMA A/C access
#define LN_EPS 1e-5f

typedef float v2f __attribute__((ext_vector_type(2)));
typedef float v8f __attribute__((ext_vector_type(8)));

struct C2 { float r, i; };
__device__ __forceinline__ C2 cmul(C2 a, C2 b) { return {a.r*b.r - a.i*b.i, a.r*b.i + a.i*b.r}; }
__device__ __forceinline__ C2 cadd(C2 a, C2 b) { return {a.r + b.r, a.i + b.i}; }

__device__ __forceinline__ float red16(float s) {
    s += __shfl_xor(s, 1); s += __shfl_xor(s, 2);
    s += __shfl_xor(s, 4); s += __shfl_xor(s, 8);
    return s;
}
__device__ __forceinline__ float red32(float s) {
    s = red16(s); s += __shfl_xor(s, 16);
    return s;
}

__global__ __launch_bounds__(256) void qdeltanet_kernel(
    const float* __restrict__ x,
    const float* __restrict__ fw,
    const float* __restrict__ ewq, const float* __restrict__ ebq,
    const float* __restrict__ ewk, const float* __restrict__ ebk,
    const float* __restrict__ ewv, const float* __restrict__ ebv,
    const float* __restrict__ xwq, const float* __restrict__ xbq,
    const float* __restrict__ xwk, const float* __restrict__ xbk,
    const float* __restrict__ xwv, const float* __restrict__ xbv,
    const float* __restrict__ bw,  const float* __restrict__ bb,
    const float* __restrict__ vqcq, const float* __restrict__ vqck, const float* __restrict__ vqcv,
    const float* __restrict__ lng, const float* __restrict__ lnb,
    float* __restrict__ outs, float* __restrict__ hout)
{
    __shared__ __align__(16) float sS[HIDDEN * SROW];     // state matrix (true S), padded rows
    __shared__ __align__(16) float sEW[3 * 4 * CTXD];     // entry weights (q,k,v)
    __shared__ __align__(16) float sEB[3 * 4];
    __shared__ __align__(16) float sXW[3 * HIDDEN * 4];   // exit weights
    __shared__ __align__(16) float sXB[3 * HIDDEN];
    __shared__ __align__(16) float sBW[INPUTD];
    __shared__ __align__(16) float sLG[HIDDEN];
    __shared__ __align__(16) float sLB[HIDDEN];
    __shared__ __align__(16) float sScale[INPUTD];        // exp(fw)
    __shared__ __align__(16) float sU[96];                // fixed VQC unitaries
    __shared__ __align__(16) float sH[HIDDEN];
    __shared__ __align__(16) float sQ[HIDDEN];            // raw (un-normalized) q
    __shared__ __align__(16) float sK[HIDDEN];            // raw (un-normalized) k
    __shared__ __align__(16) float sV[HIDDEN];
    __shared__ __align__(16) float sD[HIDDEN];            // delta pre-scaled by 1/||k||
    __shared__ __align__(16) float sZero[HIDDEN];
    __shared__ __align__(16) float sYZ[12];               // y + <Z> per branch
    __shared__ float sRed[8];                             // 0:|q|^2 1:|k|^2 2:q.k 3:sum h 4:sum h^2
    __shared__ float sBeta;
    __shared__ float sBB;

    const int tid  = threadIdx.x;
    const int bId  = blockIdx.x;
    const int lane = tid & 31;
    const int wv   = tid >> 5;
    const int hi   = (lane >> 4);
    const int n16  = lane & 15;

    // ---------------- one-time setup ----------------
    for (int i = tid; i < 768; i += 256) { sEW[i] = ewq[i]; sEW[768 + i] = ewk[i]; sEW[1536 + i] = ewv[i]; }
    if (tid < 4)   { sEB[tid] = ebq[tid]; sEB[4 + tid] = ebk[tid]; sEB[8 + tid] = ebv[tid]; }
    for (int i = tid; i < 512; i += 256) { sXW[i] = xwq[i]; sXW[512 + i] = xwk[i]; sXW[1024 + i] = xwv[i]; }
    if (tid < 128) {
        sXB[tid] = xbq[tid]; sXB[128 + tid] = xbk[tid]; sXB[256 + tid] = xbv[tid];
        sLG[tid] = lng[tid]; sLB[tid] = lnb[tid];
        sH[tid] = 0.f; sZero[tid] = 0.f;
    }
    if (tid < 64) { sBW[tid] = bw[tid]; sScale[tid] = __expf(fw[tid]); }
    if (tid == 0) { sBB = bb[0]; }
    if (tid < 8)  sRed[tid] = 0.f;
    for (int i = tid; i < HIDDEN * SROW; i += 256) sS[i] = 0.f;
    if (tid < 12) {
        // fixed per-qubit unitary U = Rz(w2) * Ry(w1) * Rx(w0)
        int br = tid >> 2, qb = tid & 3;
        const float* vw = (br == 0) ? vqcq : (br == 1) ? vqck : vqcv;
        float w0 = vw[qb], w1 = vw[4 + qb], w2 = vw[8 + qb];
        float cx, sx, cy, sy, cz, sz;
        __sincosf(0.5f * w0, &sx, &cx);
        __sincosf(0.5f * w1, &sy, &cy);
        __sincosf(0.5f * w2, &sz, &cz);
        C2 rx00 = {cx, 0.f}, rx01 = {0.f, -sx}, rx10 = {0.f, -sx}, rx11 = {cx, 0.f};
        C2 ry00 = {cy, 0.f}, ry01 = {-sy, 0.f}, ry10 = {sy, 0.f},  ry11 = {cy, 0.f};
        C2 m00 = cadd(cmul(ry00, rx00), cmul(ry01, rx10));
        C2 m01 = cadd(cmul(ry00, rx01), cmul(ry01, rx11));
        C2 m10 = cadd(cmul(ry10, rx00), cmul(ry11, rx10));
        C2 m11 = cadd(cmul(ry10, rx01), cmul(ry11, rx11));
        C2 e0 = {cz, -sz}, e1 = {cz, sz};
        C2 u00 = cmul(e0, m00), u01 = cmul(e0, m01);
        C2 u10 = cmul(e1, m10), u11 = cmul(e1, m11);
        float* up = &sU[tid * 8];
        up[0] = u00.r; up[1] = u00.i; up[2] = u01.r; up[3] = u01.i;
        up[4] = u10.r; up[5] = u10.i; up[6] = u11.r; up[7] = u11.i;
    }
    __syncthreads();

    // ---------------- sequential scan: 4 barriers per step ----------------
    for (int t = 0; t < Tsz; ++t) {
        const size_t xoff = ((size_t)bId * Tsz + t) * INPUTD;

        // ===== Phase 1: entry projections + VQC (tid 0..47), beta (48..63) =====
        if (tid < 48) {
            int br = tid >> 4;
            int n  = n16;
            // 12 combined-input values for this lane (x straight from global, h from LDS)
            float cv[12];
            #pragma unroll
            for (int u = 0; u < 12; ++u) {
                int c  = n * 12 + u;
                float xv = x[xoff + (c & 63)] * sScale[c & 63];   // clamped: never OOB
                float hv = sH[(c - INPUTD) & 127];
                cv[u] = (c < INPUTD) ? xv : hv;
            }
            float yv[4];
            #pragma unroll
            for (int j = 0; j < 4; ++j) {
                const float* w = &sEW[(br * 4 + j) * CTXD + n * 12];
                float s = 0.f;
                #pragma unroll
                for (int u = 0; u < 12; ++u) s += w[u] * cv[u];
                s = red16(s);                      // all 16 lanes now hold the full dot
                yv[j] = s + sEB[br * 4 + j];
            }
            // ---- 4-qubit VQC, one amplitude per lane, y in registers ----
            const float RS2 = 0.70710678118654752f;
            float p0r[4], p0i[4], p1r[4], p1i[4];
            #pragma unroll
            for (int w = 0; w < 4; ++w) {
                float y  = yv[w];
                float a  = atanf(y);
                float bz = atanf(y * y);
                float ca, sa, cb, sb;
                __sincosf(0.5f * a, &sa, &ca);
                __sincosf(0.5f * bz, &sb, &cb);
                float r0 = (ca - sa) * RS2;
                float r1 = (ca + sa) * RS2;
                p0r[w] = r0 * cb; p0i[w] = -r0 * sb;
                p1r[w] = r1 * cb; p1i[w] =  r1 * sb;
            }
            // CNOT ring (0,1)(1,2)(2,3)(3,0) folded into a source-index permutation
            int m = n;
            m ^= ((m >> 0) & 1) << 3;   // cnot(3,0)
            m ^= ((m >> 1) & 1) << 0;   // cnot(2,3)
            m ^= ((m >> 2) & 1) << 1;   // cnot(1,2)
            m ^= ((m >> 3) & 1) << 2;   // cnot(0,1)
            float re = 1.f, im = 0.f;
            #pragma unroll
            for (int w = 0; w < 4; ++w) {
                int bit = (m >> (3 - w)) & 1;
                float fr = bit ? p1r[w] : p0r[w];
                float fi = bit ? p1i[w] : p0i[w];
                float nr = re * fr - im * fi;
                float ni = re * fi + im * fr;
                re = nr; im = ni;
            }
            // fixed Rx*Ry*Rz per qubit via pair exchange
            #pragma unroll
            for (int i = 0; i < 4; ++i) {
                int mask = 8 >> i;
                int bsel = (n >> (3 - i)) & 1;
                int src  = (tid & 16) | (n ^ mask);
                float pr = __shfl(re, src);
                float pi = __shfl(im, src);
                const float* u = &sU[(br * 4 + i) * 8 + bsel * 4];
                float a0r = bsel ? pr : re, a0i = bsel ? pi : im;
                float a1r = bsel ? re : pr, a1i = bsel ? im : pi;
                float nr = u[0] * a0r - u[1] * a0i + u[2] * a1r - u[3] * a1i;
                float ni = u[0] * a0i + u[1] * a0r + u[2] * a1i + u[3] * a1r;
                re = nr; im = ni;
            }
            float p = re * re + im * im;
            float z[4];
            #pragma unroll
            for (int i = 0; i < 4; ++i) {
                float s = ((n >> (3 - i)) & 1) ? -p : p;
                z[i] = red16(s);
            }
            if (n == 0) {
                sYZ[br * 4 + 0] = yv[0] + z[0];
                sYZ[br * 4 + 1] = yv[1] + z[1];
                sYZ[br * 4 + 2] = yv[2] + z[2];
                sYZ[br * 4 + 3] = yv[3] + z[3];
            }
        } else if (tid < 64) {
            int lu = tid - 48;
            int c0 = lu * 4;
            float s = 0.f;
            #pragma unroll
            for (int u = 0; u < 4; ++u) s += sBW[c0 + u] * x[xoff + c0 + u] * sScale[c0 + u];
            s = red16(s);
            if (tid == 48) sBeta = 1.f / (1.f + __expf(-(s + sBB)));
            if (lu < 4 && t + 1 < Tsz) __builtin_prefetch(&x[xoff + INPUTD + lu * 16], 0, 3);
        }
        if (tid >= 248) sRed[tid - 248] = 0.f;
        __syncthreads();

        // ===== Phase 2: exit projections (raw q,k,v) + norm/dot reductions =====
        if (tid < 128) {
            int i = tid;
            float a0 = sYZ[0], a1 = sYZ[1], a2 = sYZ[2],  a3 = sYZ[3];
            float b0 = sYZ[4], b1 = sYZ[5], b2 = sYZ[6],  b3 = sYZ[7];
            float c0 = sYZ[8], c1 = sYZ[9], c2 = sYZ[10], c3 = sYZ[11];
            const float* wq = &sXW[i * 4];
            const float* wk = &sXW[512 + i * 4];
            const float* wu = &sXW[1024 + i * 4];
            float q = wq[0] * a0 + wq[1] * a1 + wq[2] * a2 + wq[3] * a3 + sXB[i];
            float k = wk[0] * b0 + wk[1] * b1 + wk[2] * b2 + wk[3] * b3 + sXB[128 + i];
            float v = wu[0] * c0 + wu[1] * c1 + wu[2] * c2 + wu[3] * c3 + sXB[256 + i];
            sQ[i] = q; sK[i] = k; sV[i] = v;
            float rq = red32(q * q);
            float rk = red32(k * k);
            float rd = red32(q * k);
            if (lane == 0) {
                atomicAdd(&sRed[0], rq);
                atomicAdd(&sRed[1], rk);
                atomicAdd(&sRed[2], rd);
            }
        }
        __syncthreads();

        // ===== Phase 3: WMMA matvec pair S x [k_raw | q_raw] + delta + h + stats =====
        {
            int row = wv * 16 + n16;
            int kv  = hi * 2;
            const float* bbase = (n16 == 0) ? sK : (n16 == 1) ? sQ : sZero;
            const float* arow  = &sS[row * SROW + kv];
            const float* brow  = bbase + kv;
            v8f acc = {0.f, 0.f, 0.f, 0.f, 0.f, 0.f, 0.f, 0.f};
            #pragma unroll
            for (int kk = 0; kk < HIDDEN; kk += 4) {
                v2f a  = *(const v2f*)(arow + kk);
                v2f bv = *(const v2f*)(brow + kk);
                acc = __builtin_amdgcn_wmma_f32_16x16x4_f32(false, a, false, bv,
                                                            (short)0, acc, false, false);
            }
            // fold l2-normalization scalars in post-processing
            float invq = 1.f / fmaxf(sqrtf(sRed[0]), 1e-12f);
            float invk = 1.f / fmaxf(sqrtf(sRed[1]), 1e-12f);
            float kq   = sRed[2] * invq * invk;       // khat . qhat
            float beta = sBeta;
            int rb = wv * 16 + 8 * hi;
            float darr[8] = {0.f, 0.f, 0.f, 0.f, 0.f, 0.f, 0.f, 0.f};
            if (n16 == 0) {
                #pragma unroll
                for (int r = 0; r < 8; ++r) {
                    float vold = acc[r] * invk;        // S . khat
                    float d = beta * (sV[rb + r] - vold);
                    darr[r] = d;
                    sD[rb + r] = d * invk;             // pre-scaled for rank-1 on raw k
                }
            }
            #pragma unroll
            for (int r = 0; r < 8; ++r)                // broadcast d within half-wave
                darr[r] = __shfl(darr[r], (lane & 16));
            if (n16 == 1) {
                float s1 = 0.f, s2 = 0.f;
                #pragma unroll
                for (int r = 0; r < 8; ++r) {
                    float h = acc[r] * invq + darr[r] * kq;   // S_new . qhat
                    sH[rb + r] = h;
                    s1 += h; s2 += h * h;
                }
                atomicAdd(&sRed[3], s1);
                atomicAdd(&sRed[4], s2);
            }
        }
        __syncthreads();

        // ===== Phase 4: layernorm output + WMMA rank-1 update S += (d/||k||) k_raw^T =====
        if (tid < 128) {
            float mu  = sRed[3] * (1.f / 128.f);
            float var = sRed[4] * (1.f / 128.f) - mu * mu;
            float o = (sH[tid] - mu) * rsqrtf(var + LN_EPS) * sLG[tid] + sLB[tid];
            outs[((size_t)bId * Tsz + t) * HIDDEN + tid] = o;
        }
        {
            int strip = wv * 16;
            float dl = sD[strip + n16];
            v2f a; a.x = hi ? 0.f : dl; a.y = 0.f;          // A col0 = d strip
            #pragma unroll
            for (int ct = 0; ct < 8; ++ct) {
                float kl = sK[ct * 16 + n16];
                v2f bv; bv.x = hi ? 0.f : kl; bv.y = 0.f;   // B row0 = raw k chunk
                int col = ct * 16 + n16;
                int rb  = strip + 8 * hi;
                v8f c;
                #pragma unroll
                for (int r = 0; r < 8; ++r) c[r] = sS[(rb + r) * SROW + col];
                c = __builtin_amdgcn_wmma_f32_16x16x4_f32(false, a, false, bv,
                                                          (short)0, c, false, false);
                #pragma unroll
                for (int r = 0; r < 8; ++r) sS[(rb + r) * SROW + col] = c[r];
            }
        }
        __syncthreads();
    }

    if (tid < 128) hout[(size_t)bId * HIDDEN + tid] = sH[tid];
}

extern "C" void kernel_launch(void* const* d_in, const int* in_sizes, int n_in,
                              void* d_out, int out_size, void* d_ws, size_t ws_size,
                              hipStream_t stream) {
    (void)in_sizes; (void)n_in; (void)out_size; (void)d_ws; (void)ws_size;
    const float* x    = (const float*)d_in[0];
    const float* fw   = (const float*)d_in[1];
    const float* ewq  = (const float*)d_in[2];
    const float* ebq  = (const float*)d_in[3];
    const float* ewk  = (const float*)d_in[4];
    const float* ebk  = (const float*)d_in[5];
    const float* ewv  = (const float*)d_in[6];
    const float* ebv  = (const float*)d_in[7];
    const float* xwq  = (const float*)d_in[8];
    const float* xbq  = (const float*)d_in[9];
    const float* xwk  = (const float*)d_in[10];
    const float* xbk  = (const float*)d_in[11];
    const float* xwv  = (const float*)d_in[12];
    const float* xbv  = (const float*)d_in[13];
    const float* bw   = (const float*)d_in[14];
    const float* bb   = (const float*)d_in[15];
    const float* vqcq = (const float*)d_in[16];
    const float* vqck = (const float*)d_in[17];
    const float* vqcv = (const float*)d_in[18];
    const float* lng  = (const float*)d_in[19];
    const float* lnb  = (const float*)d_in[20];

    float* outs = (float*)d_out;
    float* hout = outs + (size_t)Bsz * Tsz * HIDDEN;

    qdeltanet_kernel<<<Bsz, 256, 0, stream>>>(
        x, fw, ewq, ebq, ewk, ebk, ewv, ebv,
        xwq, xbq, xwk, xbk, xwv, xbv, bw, bb,
        vqcq, vqck, vqcv, lng, lnb, outs, hout);
}